// NSMCell_28424093565197
// MI455X (gfx1250) — compile-verified
//
#include <hip/hip_runtime.h>
#include <math.h>

// CDNA5 / gfx1250, wave32. f32 WMMA 16x16x4 for both GEMMs.

typedef __attribute__((ext_vector_type(2))) float v2f;
typedef __attribute__((ext_vector_type(8))) float v8f;

#define HDIM 128
#define XSTR 132   // padded LDS row stride (floats): avoids bank conflicts, keeps 8B align

__device__ __forceinline__ float elu1(float x) { return x > 0.f ? x : expm1f(x); }

__device__ __forceinline__ void atomicMaxF(float* addr, float val) {
  int* ai = (int*)addr;
  int cur = __float_as_int(*addr);
  while (__int_as_float(cur) < val) {
    int assumed = cur;
    cur = atomicCAS(ai, assumed, __float_as_int(val));
    if (cur == assumed) break;
  }
}

// ---------------- K0: init msg + segment stats ----------------
__global__ void init_k(float* __restrict__ msg, unsigned long long total,
                       float* __restrict__ m_node, float* __restrict__ m_rel,
                       float* __restrict__ sum_node, float* __restrict__ sum_rel, int B) {
  unsigned long long i = (unsigned long long)blockIdx.x * blockDim.x + threadIdx.x;
  if (i < total) msg[i] = 0.f;
  if (blockIdx.x == 0 && (int)threadIdx.x < B) {
    m_node[threadIdx.x] = -INFINITY;
    m_rel[threadIdx.x]  = -INFINITY;
    sum_node[threadIdx.x] = 0.f;
    sum_rel[threadIdx.x]  = 0.f;
  }
}

// ---------------- K1: node scores -> s_node[n] ----------------
// s_node[n] = dot(elu(sum_p sim[b,p]*(instr[b] .* attrs[n,p]) @ W_props[p]), w_node_score)
__global__ __launch_bounds__(128) void node_kernel(
    const float* __restrict__ node_attrs, const float* __restrict__ instr,
    const float* __restrict__ nps, const float* __restrict__ W_props,
    const float* __restrict__ wns, const int* __restrict__ nidx,
    float* __restrict__ s_node, int N, int P) {
  __shared__ float Xs[4][16 * XSTR];
  const int tid  = threadIdx.x;
  const int w    = tid >> 5;
  const int lane = tid & 31;
  const int idx  = lane & 15;
  const int half = lane >> 4;
  const int kh   = 2 * half;
  const int n0   = blockIdx.x * 64 + w * 16;
  const int c    = lane * 4;

  v8f acc[8];
#pragma unroll
  for (int t = 0; t < 8; ++t) acc[t] = (v8f){0.f,0.f,0.f,0.f,0.f,0.f,0.f,0.f};

  for (int p = 0; p < P; ++p) {
    // stage gated tile: X[r][h] = sim[b,p] * instr[b][h] * attrs[n,p,h]
    for (int r = 0; r < 16; ++r) {
      int n = n0 + r;
      float x0 = 0.f, x1 = 0.f, x2 = 0.f, x3 = 0.f;
      if (n < N) {
        int b = nidx[n];
        float sv = nps[b * P + p];
        const float4 at = *(const float4*)(node_attrs + ((size_t)n * P + p) * HDIM + c);
        const float4 ib = *(const float4*)(instr + (size_t)b * HDIM + c);
        x0 = sv * at.x * ib.x; x1 = sv * at.y * ib.y;
        x2 = sv * at.z * ib.z; x3 = sv * at.w * ib.w;
      }
      float* row = &Xs[w][r * XSTR + c];
      row[0] = x0; row[1] = x1; row[2] = x2; row[3] = x3;
    }
    __syncthreads();
    const float* Wp = W_props + (size_t)p * HDIM * HDIM;
#pragma unroll
    for (int t = 0; t < 8; ++t) {
      const int n = t * 16 + idx;
#pragma unroll 4
      for (int k0 = 0; k0 < HDIM; k0 += 4) {
        v2f a, b;
        a.x = Xs[w][idx * XSTR + k0 + kh];
        a.y = Xs[w][idx * XSTR + k0 + kh + 1];
        b.x = Wp[(k0 + kh) * HDIM + n];
        b.y = Wp[(k0 + kh + 1) * HDIM + n];
        acc[t] = __builtin_amdgcn_wmma_f32_16x16x4_f32(
            false, a, false, b, (short)0, acc[t], false, false);
      }
    }
    __syncthreads();
  }

  // elu + dot with w_node_score, reduce across the 16 lanes of each half
  float part[8];
#pragma unroll
  for (int i = 0; i < 8; ++i) part[i] = 0.f;
#pragma unroll
  for (int t = 0; t < 8; ++t) {
    float wv = wns[t * 16 + idx];
#pragma unroll
    for (int i = 0; i < 8; ++i) part[i] += elu1(acc[t][i]) * wv;
  }
#pragma unroll
  for (int mask = 1; mask <= 8; mask <<= 1) {
#pragma unroll
    for (int i = 0; i < 8; ++i) part[i] += __shfl_xor(part[i], mask, 32);
  }
  if (idx == 0) {
#pragma unroll
    for (int i = 0; i < 8; ++i) {
      int n = n0 + i + 8 * half;
      if (n < N) s_node[n] = part[i];
    }
  }
}

// ---------------- K2: edge GEMM + ELU + scatter into msg ----------------
__global__ __launch_bounds__(128) void edge_kernel(
    const float* __restrict__ edge_attrs, const float* __restrict__ instr,
    const float* __restrict__ dist, const float* __restrict__ W_edge,
    const int* __restrict__ eidx, const int* __restrict__ ebi,
    float* __restrict__ msg, int E) {
  __shared__ float Xs[4][16 * XSTR];
  __shared__ int   dsts[64];
  __shared__ float scales[64];
  const int tid  = threadIdx.x;
  const int w    = tid >> 5;
  const int lane = tid & 31;
  const int idx  = lane & 15;
  const int half = lane >> 4;
  const int kh   = 2 * half;
  const int e0   = blockIdx.x * 64 + w * 16;
  const int c    = lane * 4;

  if (lane < 16) {
    int e = e0 + lane, d = 0; float sc = 0.f;
    if (e < E) { d = eidx[(size_t)E + e]; sc = dist[eidx[e]]; }
    dsts[w * 16 + lane] = d;
    scales[w * 16 + lane] = sc;
  }
  for (int r = 0; r < 16; ++r) {
    int e = e0 + r;
    float x0 = 0.f, x1 = 0.f, x2 = 0.f, x3 = 0.f;
    if (e < E) {
      int b = ebi[e];
      const float4 ea = *(const float4*)(edge_attrs + (size_t)e * HDIM + c);
      const float4 ib = *(const float4*)(instr + (size_t)b * HDIM + c);
      x0 = ea.x * ib.x; x1 = ea.y * ib.y; x2 = ea.z * ib.z; x3 = ea.w * ib.w;
    }
    float* row = &Xs[w][r * XSTR + c];
    row[0] = x0; row[1] = x1; row[2] = x2; row[3] = x3;
  }
  __syncthreads();

#pragma unroll 1
  for (int t = 0; t < 8; ++t) {
    v8f acc = (v8f){0.f,0.f,0.f,0.f,0.f,0.f,0.f,0.f};
    const int n = t * 16 + idx;
#pragma unroll 4
    for (int k0 = 0; k0 < HDIM; k0 += 4) {
      v2f a, b;
      a.x = Xs[w][idx * XSTR + k0 + kh];
      a.y = Xs[w][idx * XSTR + k0 + kh + 1];
      b.x = W_edge[(k0 + kh) * HDIM + n];
      b.y = W_edge[(k0 + kh + 1) * HDIM + n];
      acc = __builtin_amdgcn_wmma_f32_16x16x4_f32(
          false, a, false, b, (short)0, acc, false, false);
    }
#pragma unroll
    for (int i = 0; i < 8; ++i) {
      int m = i + 8 * half;
      int e = e0 + m;
      if (e < E) {
        float v = elu1(acc[i]) * scales[w * 16 + m];
        atomicAdd(msg + (size_t)dsts[w * 16 + m] * HDIM + n, v);
      }
    }
  }
}

// ---------------- K3: s_rel[n] = dot(msg[n], w_rel_score) ----------------
__global__ void rowdot_k(const float* __restrict__ msg, const float* __restrict__ wrs,
                         float* __restrict__ s_rel, int N) {
  const int lane = threadIdx.x & 31;
  const int wid  = (blockIdx.x * blockDim.x + threadIdx.x) >> 5;
  const int nw   = (gridDim.x * blockDim.x) >> 5;
  const float4 w4 = *(const float4*)(wrs + lane * 4);
  for (int n = wid; n < N; n += nw) {
    const float4 m4 = *(const float4*)(msg + (size_t)n * HDIM + lane * 4);
    float v = m4.x * w4.x + m4.y * w4.y + m4.z * w4.z + m4.w * w4.w;
#pragma unroll
    for (int mask = 16; mask >= 1; mask >>= 1) v += __shfl_xor(v, mask, 32);
    if (lane == 0) s_rel[n] = v;
  }
}

// ---------------- K4: per-segment max ----------------
__global__ void seg_max_k(const float* __restrict__ s_node, const float* __restrict__ s_rel,
                          const int* __restrict__ nidx, float* __restrict__ m_node,
                          float* __restrict__ m_rel, int N, int B) {
  __shared__ float lmn[64], lmr[64];
  const int tid = threadIdx.x;
  if (tid < B) { lmn[tid] = -INFINITY; lmr[tid] = -INFINITY; }
  __syncthreads();
  for (int n = blockIdx.x * blockDim.x + tid; n < N; n += gridDim.x * blockDim.x) {
    int b = nidx[n];
    atomicMaxF(&lmn[b], s_node[n]);
    atomicMaxF(&lmr[b], s_rel[n]);
  }
  __syncthreads();
  if (tid < B) { atomicMaxF(&m_node[tid], lmn[tid]); atomicMaxF(&m_rel[tid], lmr[tid]); }
}

// ---------------- K5: per-segment exp-sum ----------------
__global__ void seg_sum_k(const float* __restrict__ s_node, const float* __restrict__ s_rel,
                          const int* __restrict__ nidx, const float* __restrict__ m_node,
                          const float* __restrict__ m_rel, float* __restrict__ sum_node,
                          float* __restrict__ sum_rel, int N, int B) {
  __shared__ float lsn[64], lsr[64];
  const int tid = threadIdx.x;
  if (tid < B) { lsn[tid] = 0.f; lsr[tid] = 0.f; }
  __syncthreads();
  for (int n = blockIdx.x * blockDim.x + tid; n < N; n += gridDim.x * blockDim.x) {
    int b = nidx[n];
    atomicAdd(&lsn[b], expf(s_node[n] - m_node[b]));
    atomicAdd(&lsr[b], expf(s_rel[n] - m_rel[b]));
  }
  __syncthreads();
  if (tid < B) { atomicAdd(&sum_node[tid], lsn[tid]); atomicAdd(&sum_rel[tid], lsr[tid]); }
}

// ---------------- K6: blend ----------------
__global__ void finalize_k(const float* __restrict__ s_node, const float* __restrict__ s_rel,
                           const int* __restrict__ nidx, const float* __restrict__ rels,
                           const float* __restrict__ m_node, const float* __restrict__ m_rel,
                           const float* __restrict__ sum_node, const float* __restrict__ sum_rel,
                           float* __restrict__ out, int N) {
  int n = blockIdx.x * blockDim.x + threadIdx.x;
  if (n >= N) return;
  int b = nidx[n];
  float rs = rels[b];
  float pn = expf(s_node[n] - m_node[b]) / sum_node[b];
  float pr = expf(s_rel[n]  - m_rel[b])  / sum_rel[b];
  out[n] = rs * pr + (1.f - rs) * pn;
}

extern "C" void kernel_launch(void* const* d_in, const int* in_sizes, int n_in,
                              void* d_out, int out_size, void* d_ws, size_t ws_size,
                              hipStream_t stream) {
  const float* node_attrs = (const float*)d_in[0];
  const float* edge_attrs = (const float*)d_in[1];
  const float* instr      = (const float*)d_in[2];
  const float* dist       = (const float*)d_in[3];
  const float* nps        = (const float*)d_in[4];
  const float* rels       = (const float*)d_in[5];
  const float* W_props    = (const float*)d_in[6];
  const float* W_edge     = (const float*)d_in[7];
  const float* wns        = (const float*)d_in[8];
  const float* wrs        = (const float*)d_in[9];
  const int*   eidx       = (const int*)d_in[10];
  const int*   nidx       = (const int*)d_in[11];
  const int*   ebi        = (const int*)d_in[12];

  const int N = in_sizes[3];          // distribution: (N,)
  const int B = in_sizes[5];          // relation_similarity: (B,)
  const int E = in_sizes[12];         // edge_batch_indices: (E,)
  const int P = in_sizes[4] / B;      // node_prop_similarities: (B,P)

  float* msg      = (float*)d_ws;                       // N*HDIM
  float* s_node   = msg + (size_t)N * HDIM;             // N
  float* s_rel    = s_node + N;                         // N
  float* m_node   = s_rel + N;                          // B
  float* m_rel    = m_node + B;
  float* sum_node = m_rel + B;
  float* sum_rel  = sum_node + B;
  float* out      = (float*)d_out;

  unsigned long long total = (unsigned long long)N * HDIM;
  int initBlocks = (int)((total + 255) / 256);
  init_k<<<initBlocks, 256, 0, stream>>>(msg, total, m_node, m_rel, sum_node, sum_rel, B);

  int nodeBlocks = (N + 63) / 64;
  node_kernel<<<nodeBlocks, 128, 0, stream>>>(node_attrs, instr, nps, W_props, wns, nidx,
                                              s_node, N, P);

  int edgeBlocks = (E + 63) / 64;
  edge_kernel<<<edgeBlocks, 128, 0, stream>>>(edge_attrs, instr, dist, W_edge, eidx, ebi,
                                              msg, E);

  int dotBlocks = (N + 7) / 8;        // 8 waves per 256-thread block, one row per wave
  rowdot_k<<<dotBlocks, 256, 0, stream>>>(msg, wrs, s_rel, N);

  seg_max_k<<<512, 256, 0, stream>>>(s_node, s_rel, nidx, m_node, m_rel, N, B);
  seg_sum_k<<<512, 256, 0, stream>>>(s_node, s_rel, nidx, m_node, m_rel, sum_node, sum_rel, N, B);

  finalize_k<<<(N + 255) / 256, 256, 0, stream>>>(s_node, s_rel, nidx, rels, m_node, m_rel,
                                                  sum_node, sum_rel, out, N);
}